// GCN_minibatch_42021960024582
// MI455X (gfx1250) — compile-verified
//
#include <hip/hip_runtime.h>
#include <hip/hip_bf16.h>

typedef float v2f __attribute__((ext_vector_type(2)));
typedef float v8f __attribute__((ext_vector_type(8)));

// ---------------------------------------------------------------------------
// Guaranteed-native fp32 atomic add (no return -> STOREcnt; S_ENDPGM's
// implicit wait-idle covers completion).
// ---------------------------------------------------------------------------
__device__ __forceinline__ void gadd_f32(float* p, float v) {
    asm volatile("global_atomic_add_f32 %0, %1, off" : : "v"(p), "v"(v) : "memory");
}

// ---------------------------------------------------------------------------
// Zero-fill kernel (accumulator init; float4 stores)
// ---------------------------------------------------------------------------
__global__ void gcn_zero_kernel(float* __restrict__ p, size_t n_floats) {
    size_t i = ((size_t)blockIdx.x * blockDim.x + threadIdx.x) * 4;
    if (i + 3 < n_floats) {
        *(float4*)(p + i) = make_float4(0.f, 0.f, 0.f, 0.f);
    } else {
        for (size_t j = i; j < n_floats; ++j) p[j] = 0.f;
    }
}

// ---------------------------------------------------------------------------
// Edge aggregation: agg[dst[e], :] += feat[src[e], :]
// One wave per edge; lane handles 4 contiguous floats of the 128-float row:
// one b128 gather + 4 global_atomic_add_f32. Feature table (51 MB) and
// accumulator (51 MB) both fit MI455X's 192 MB L2 -> L2-atomic-bound.
// ---------------------------------------------------------------------------
__global__ void gcn_aggregate_kernel(const float* __restrict__ feat,
                                     const int* __restrict__ src,
                                     const int* __restrict__ dst,
                                     float* __restrict__ agg,
                                     int n_edges) {
    int edge = (int)(((size_t)blockIdx.x * blockDim.x + threadIdx.x) >> 5);
    int lane = threadIdx.x & 31;
    if (edge >= n_edges) return;

    int s = __builtin_amdgcn_readfirstlane(src[edge]);
    int d = __builtin_amdgcn_readfirstlane(dst[edge]);

    const float4 v = *(const float4*)(feat + (size_t)s * 128 + lane * 4);
    float* out = agg + (size_t)d * 128 + lane * 4;
    gadd_f32(out + 0, v.x);
    gadd_f32(out + 1, v.y);
    gadd_f32(out + 2, v.z);
    gadd_f32(out + 3, v.w);
}

// ---------------------------------------------------------------------------
// GEMM: C[n_rows, n_cols] = A[n_rows, 128] @ W[128, n_cols] + bias (+ReLU)
// Full fp32 via V_WMMA_F32_16X16X4_F32; one wave per 16x16 C tile.
// Operand striping per CDNA5 ISA 7.12.2:
//   A (16x4 f32): lane L -> row M = L%16, VGPR r -> K = r + 2*(L/16)
//   B (4x16 f32): lane L -> col N = L%16, VGPR r -> K = r + 2*(L/16)
//   C (16x16):    VGPR r -> row = r + 8*(L/16), col = L%16
// NCOLS > 0 = compile-time column count: every W load / C store offset is a
// 24-bit instruction immediate, so the fully-unrolled K loop carries zero
// address arithmetic. Out-of-range rows/cols are index-CLAMPED (loads stay
// in-bounds, EXEC stays all-ones for WMMA); only stores are masked, with one
// tile-uniform row branch and one per-lane column branch around all 8 stores.
// ---------------------------------------------------------------------------
template <bool RELU, int NCOLS>
__global__ void gcn_gemm_kernel(const float* __restrict__ A,
                                const float* __restrict__ W,
                                const float* __restrict__ bias,
                                float* __restrict__ C,
                                int n_rows, int n_cols_rt) {
    const int n_cols = (NCOLS > 0) ? NCOLS : n_cols_rt;

    const int wave = threadIdx.x >> 5;
    const int lane = threadIdx.x & 31;
    const int half = lane >> 4;          // 0: lanes 0-15, 1: lanes 16-31
    const int l16  = lane & 15;
    const int koff = half * 2;           // K offset this lane-group supplies

    const int tile_m = blockIdx.x;
    const int n   = wave * 16 + l16;                 // C/W column
    const int m   = tile_m * 16 + l16;               // A row
    const int ncl = (n < n_cols) ? n : (n_cols - 1); // clamped (store-masked)
    const int mcl = (m < n_rows) ? m : (n_rows - 1);

    const float* __restrict__ ap = A + mcl * 128 + koff;
    const float* __restrict__ wp = W + koff * n_cols + ncl;

    v8f acc = {};
    if (NCOLS > 0) {
        // Compile-time strides: immediate-offset loads, no loop arithmetic.
#pragma unroll
        for (int k0 = 0; k0 < 128; k0 += 4) {
            v2f a = *(const v2f*)(ap + k0);          // one b64, offset:imm
            v2f b;
            b.x = wp[k0 * NCOLS];                    // b32, offset:imm
            b.y = wp[(k0 + 1) * NCOLS];              // b32, offset:imm
            acc = __builtin_amdgcn_wmma_f32_16x16x4_f32(
                false, a, false, b, (short)0, acc, false, false);
        }
    } else {
        const int wstep = 4 * n_cols;
#pragma unroll 8
        for (int k0 = 0; k0 < 128; k0 += 4) {
            v2f a = *(const v2f*)ap;
            v2f b;
            b.x = wp[0];
            b.y = wp[n_cols];
            acc = __builtin_amdgcn_wmma_f32_16x16x4_f32(
                false, a, false, b, (short)0, acc, false, false);
            ap += 4;
            wp += wstep;
        }
    }

    const bool nvalid = (n < n_cols);
    const float bn = bias[ncl];
    const int row0 = tile_m * 16 + half * 8;
    float* __restrict__ cp = C + (size_t)row0 * n_cols + n;

    if (tile_m * 16 + 15 < n_rows) {                 // tile-uniform branch
        if (nvalid) {                                // single saveexec
#pragma unroll
            for (int r = 0; r < 8; ++r) {
                float v = acc[r] + bn;
                if (RELU) v = fmaxf(v, 0.f);
                cp[r * n_cols] = v;                  // store, offset:imm
            }
        }
    } else {                                         // ragged last tile
#pragma unroll
        for (int r = 0; r < 8; ++r) {
            if (nvalid && (row0 + r) < n_rows) {
                float v = acc[r] + bn;
                if (RELU) v = fmaxf(v, 0.f);
                cp[r * n_cols] = v;
            }
        }
    }
}

// ---------------------------------------------------------------------------
// Launch: 3x { zero, aggregate, gemm }
// ---------------------------------------------------------------------------
extern "C" void kernel_launch(void* const* d_in, const int* in_sizes, int n_in,
                              void* d_out, int out_size, void* d_ws, size_t ws_size,
                              hipStream_t stream) {
    const float* features = (const float*)d_in[0];
    const int*   src      = (const int*)d_in[1];
    const int*   dst      = (const int*)d_in[2];
    const float* W0       = (const float*)d_in[3];
    const float* b0       = (const float*)d_in[4];
    const float* W1       = (const float*)d_in[5];
    const float* b1       = (const float*)d_in[6];
    const float* W2       = (const float*)d_in[7];
    const float* b2       = (const float*)d_in[8];
    float*       out      = (float*)d_out;

    const int feat_dim  = 128;                       // IN_FEATS == N_HIDDEN
    const int n_nodes   = in_sizes[0] / feat_dim;    // 100000
    const int n_edges   = in_sizes[1];               // 1.6M
    const int n_classes = in_sizes[8];               // 40

    float* agg = (float*)d_ws;                                   // [N,128]
    float* h   = agg + (size_t)n_nodes * feat_dim;               // [N,128]

    const size_t nfloats = (size_t)n_nodes * feat_dim;
    const int zblocks = (int)((nfloats / 4 + 255) / 256);
    const int eblocks = (n_edges + 7) / 8;           // 8 edges / 256-thr block
    const int mtiles  = (n_nodes + 15) / 16;         // 6250 (exact)
    const int tiles_hidden = feat_dim / 16;          // 8 waves
    const int tiles_out    = (n_classes + 15) / 16;  // 3 waves

    // Layer 0: agg = scatter(features); h = relu(agg@W0 + b0)
    gcn_zero_kernel<<<zblocks, 256, 0, stream>>>(agg, nfloats);
    gcn_aggregate_kernel<<<eblocks, 256, 0, stream>>>(features, src, dst, agg, n_edges);
    gcn_gemm_kernel<true, 128><<<mtiles, tiles_hidden * 32, 0, stream>>>(
        agg, W0, b0, h, n_nodes, feat_dim);

    // Layer 1: agg = scatter(h); h = relu(agg@W1 + b1)
    gcn_zero_kernel<<<zblocks, 256, 0, stream>>>(agg, nfloats);
    gcn_aggregate_kernel<<<eblocks, 256, 0, stream>>>(h, src, dst, agg, n_edges);
    gcn_gemm_kernel<true, 128><<<mtiles, tiles_hidden * 32, 0, stream>>>(
        agg, W1, b1, h, n_nodes, feat_dim);

    // Layer 2: agg = scatter(h); out = agg@W2 + b2
    gcn_zero_kernel<<<zblocks, 256, 0, stream>>>(agg, nfloats);
    gcn_aggregate_kernel<<<eblocks, 256, 0, stream>>>(h, src, dst, agg, n_edges);
    if (n_classes == 40) {
        gcn_gemm_kernel<false, 40><<<mtiles, tiles_out * 32, 0, stream>>>(
            agg, W2, b2, out, n_nodes, n_classes);
    } else {
        gcn_gemm_kernel<false, 0><<<mtiles, tiles_out * 32, 0, stream>>>(
            agg, W2, b2, out, n_nodes, n_classes);
    }
}